// LocalAttention_36240934043916
// MI455X (gfx1250) — compile-verified
//
#include <hip/hip_runtime.h>
#include <stdint.h>

// LocalAttention r=3 on MI455X (gfx1250, wave32).
// logits via V_WMMA_F32_16X16X4_F32 banded GEMM; key tile staged to LDS with
// gfx1250 async-to-LDS copies (ASYNCcnt); softmax + weighted-sum on VALU.

typedef __attribute__((ext_vector_type(2))) float v2f;
typedef __attribute__((ext_vector_type(8))) float v8f;
typedef int v4i32 __attribute__((vector_size(16)));

#define C_DIM 256
#define H_DIM 128
#define W_DIM 128
#define TW    16            // pixels per tile (one row segment)
#define NT    (W_DIM / TW)  // 8 tiles per row
// padded key-tile columns: 0..15 = w0..w0+15 (16B-aligned interior),
// 16 = w0-1 (left halo), 17 = w0+16 (right halo), 18..19 = pad (never read)
#define KC_P  20

#if __has_builtin(__builtin_amdgcn_global_load_async_to_lds_b128) && \
    __has_builtin(__builtin_amdgcn_global_load_async_to_lds_b32)
#define USE_ASYNC_LDS 1
#else
#define USE_ASYNC_LDS 0
#endif

#define AS1 __attribute__((address_space(1)))
#define AS3 __attribute__((address_space(3)))

__global__ __launch_bounds__(256) void local_attn_wmma_kernel(
    const float* __restrict__ key,
    const float* __restrict__ query,
    float* __restrict__ out)
{
    __shared__ __align__(16) float k_lds[3][C_DIM][KC_P];  // 61440 B
    __shared__ float logits[TW][16];  // [pixel][k]; col 15 = write-dump, 9..14 unused

    const int tile = blockIdx.x;            // n*H*NT + h*NT + wt
    const int wt = tile % NT;
    const int h  = (tile / NT) % H_DIM;
    const int n  = tile / (NT * H_DIM);
    const int w0 = wt * TW;
    const int t  = threadIdx.x;

    // ================= stage key tile (3 rows x 256 ch x 18 cols) =================
#if USE_ASYNC_LDS
    // Pre-zero only out-of-bounds halo/row slots (disjoint from async targets).
    const bool edge = (h == 0) | (h == H_DIM - 1) | (wt == 0) | (wt == NT - 1);
    if (edge) {
        for (int i = t; i < 3 * C_DIM * 18; i += 256) {
            const int c2 = i % 18;
            const int ch = (i / 18) % C_DIM;
            const int kh = i / (18 * C_DIM);
            const int gw = (c2 == 16) ? (w0 - 1) : ((c2 == 17) ? (w0 + 16) : (w0 + c2));
            const int gh = h - 1 + kh;
            if ((unsigned)gw >= (unsigned)W_DIM || (unsigned)gh >= (unsigned)H_DIM)
                k_lds[kh][ch][c2] = 0.0f;
        }
    }
    // Interior: 3*256 rows x 16 cols as aligned async B128 copies (12 per thread).
    for (int i = t; i < 3 * C_DIM * 4; i += 256) {
        const int qv = i & 3;
        const int ch = (i >> 2) % C_DIM;
        const int kh = i / (4 * C_DIM);
        const int gh = h - 1 + kh;
        if ((unsigned)gh < (unsigned)H_DIM) {
            const float* gp = key + (((size_t)n * C_DIM + ch) * H_DIM + gh) * W_DIM + w0 + 4 * qv;
            __builtin_amdgcn_global_load_async_to_lds_b128(
                (AS1 v4i32*)(uintptr_t)(const void*)gp,
                (AS3 v4i32*)&k_lds[kh][ch][4 * qv], 0, 0);
        }
    }
    // Halo columns: async B32 copies where in-bounds (6 per thread).
    for (int i = t; i < 3 * C_DIM * 2; i += 256) {
        const int side = i & 1;
        const int ch   = (i >> 1) % C_DIM;
        const int kh   = i / (2 * C_DIM);
        const int gh   = h - 1 + kh;
        const int gw   = side ? (w0 + 16) : (w0 - 1);
        if ((unsigned)gh < (unsigned)H_DIM && (unsigned)gw < (unsigned)W_DIM) {
            const float* gp = key + (((size_t)n * C_DIM + ch) * H_DIM + gh) * W_DIM + gw;
            __builtin_amdgcn_global_load_async_to_lds_b32(
                (AS1 int*)(uintptr_t)(const void*)gp,
                (AS3 int*)&k_lds[kh][ch][16 + side], 0, 0);
        }
    }
#if __has_builtin(__builtin_amdgcn_s_wait_asynccnt)
    __builtin_amdgcn_s_wait_asynccnt(0);
#else
    asm volatile("s_wait_asynccnt 0x0" ::: "memory");
#endif
#else
    // Fallback: plain load + ds-store staging with zero padding.
    for (int i = t; i < 3 * C_DIM * 18; i += 256) {
        const int c2 = i % 18;
        const int ch = (i / 18) % C_DIM;
        const int kh = i / (18 * C_DIM);
        const int gw = (c2 == 16) ? (w0 - 1) : ((c2 == 17) ? (w0 + 16) : (w0 + c2));
        const int gh = h - 1 + kh;
        float v = 0.0f;
        if ((unsigned)gw < (unsigned)W_DIM && (unsigned)gh < (unsigned)H_DIM)
            v = key[(((size_t)n * C_DIM + ch) * H_DIM + gh) * W_DIM + gw];
        k_lds[kh][ch][c2] = v;
    }
#endif
    __syncthreads();

    // ========== logits: banded GEMM S = Q^T(16x256) x K_row(256x18) per kh ==========
    // 6 wave-level GEMMs: wave wv -> kh = wv/2, column half gh2 = wv%2.
    const int wv   = t >> 5;
    const int lane = t & 31;
    if (wv < 6) {
        const int kh  = wv >> 1;
        const int gh2 = wv & 1;
        const int l16 = lane & 15;
        const int lhi = lane >> 4;                 // lane-half -> K sub-pair (ISA layout)
        const int j   = gh2 * 16 - 1 + l16;        // score column rel. to w0: -1..30
        const bool jv = (j <= 16);                 // columns beyond w0+16 are dead
        // LDS column for score column j (clamped-safe for dead lanes):
        const int lcol = (j < 0) ? 16 : ((j == 16) ? 17 : ((j <= 15) ? j : 0));

        const float* qb   = query + (((size_t)n * C_DIM) * H_DIM + h) * W_DIM + w0 + l16;
        const float* krow = &k_lds[kh][0][0];

        v8f acc = {};
        #pragma unroll 8
        for (int kk = 0; kk < C_DIM / 4; ++kk) {
            const int ch = kk * 4 + lhi * 2;       // this lane-half's two K channels
            v2f a, b;
            a.x = qb[(size_t)(ch + 0) * (H_DIM * W_DIM)];
            a.y = qb[(size_t)(ch + 1) * (H_DIM * W_DIM)];
            // unconditional safe LDS reads + value select (v_cndmask, no EXEC dance)
            const float k0 = krow[(ch + 0) * KC_P + lcol];
            const float k1 = krow[(ch + 1) * KC_P + lcol];
            b.x = jv ? k0 : 0.0f;
            b.y = jv ? k1 : 0.0f;
            acc = __builtin_amdgcn_wmma_f32_16x16x4_f32(
                false, a, false, b, (short)0, acc, false, false);
        }

        // band extraction: lane holds S[p = v + 8*lhi, j].
        // Unconditional stores; out-of-band lanes dump into logits[p][15]
        // (address select instead of EXEC-masked branches; col 15 never read).
        #pragma unroll
        for (int v = 0; v < 8; ++v) {
            const int p   = v + 8 * lhi;
            const int kw  = j - p + 1;
            const int kcl = ((unsigned)kw <= 2u) ? (kh * 3 + kw) : 15;
            logits[p][kcl] = acc[v];
        }
    }
    __syncthreads();

    // ===== softmax over 9 neighbors (padded neighbors have logit 0, as reference) =====
    if (t < TW) {
        float m = logits[t][0];
        #pragma unroll
        for (int k = 1; k < 9; ++k) m = fmaxf(m, logits[t][k]);
        float e[9], s = 0.0f;
        #pragma unroll
        for (int k = 0; k < 9; ++k) { e[k] = expf(logits[t][k] - m); s += e[k]; }
        const float inv = 1.0f / s;
        #pragma unroll
        for (int k = 0; k < 9; ++k) logits[t][k] = e[k] * inv;
    }
    __syncthreads();

    // ===== out[c,p] = sum_k attn[p,k] * key_patch[c,k,p]  (VALU, LDS-resident keys) =====
    const int px = t & 15;
    const int cb = t >> 4;
    const int c0 = (px == 0)  ? 16 : (px - 1);
    const int c1 = px;
    const int c2 = (px == 15) ? 17 : (px + 1);
    float a[9];
    #pragma unroll
    for (int k = 0; k < 9; ++k) a[k] = logits[px][k];   // broadcast, conflict-free

    #pragma unroll 4
    for (int ci = 0; ci < C_DIM / 16; ++ci) {
        const int ch = ci * 16 + cb;
        float acc = 0.0f;
        #pragma unroll
        for (int kh = 0; kh < 3; ++kh) {
            acc = fmaf(a[kh * 3 + 0], k_lds[kh][ch][c0], acc);
            acc = fmaf(a[kh * 3 + 1], k_lds[kh][ch][c1], acc);
            acc = fmaf(a[kh * 3 + 2], k_lds[kh][ch][c2], acc);
        }
        out[(((size_t)n * C_DIM + ch) * H_DIM + h) * W_DIM + w0 + px] = acc;
    }
}

extern "C" void kernel_launch(void* const* d_in, const int* in_sizes, int n_in,
                              void* d_out, int out_size, void* d_ws, size_t ws_size,
                              hipStream_t stream) {
    (void)in_sizes; (void)n_in; (void)d_ws; (void)ws_size; (void)out_size;
    const float* key   = (const float*)d_in[0];
    const float* query = (const float*)d_in[1];
    float* out         = (float*)d_out;

    const int n_tiles = 4 * H_DIM * NT;   // 4096 workgroups
    local_attn_wmma_kernel<<<dim3(n_tiles), dim3(256), 0, stream>>>(key, query, out);
}